// HybridQuantumClassifier_43293270343728
// MI455X (gfx1250) — compile-verified
//
#include <hip/hip_runtime.h>

#define NQ 14
#define NL 6
#define NSTATE (1 << NQ)          // 16384 amplitudes
#define BLOCK 256
#define NWAVES (BLOCK / 32)

// LDS layout (floats):
//   [0, 32768)        state as interleaved float2 {re, im}  (16384 amplitudes)
//   [32768, 33024)    Ure 16x16
//   [33024, 33280)    Uim 16x16
//   [33280, 33536)    Uin 16x16  (= -Uim, since f32 WMMA has no A/B negate)
//   [33536, ...)      reduction scratch (NWAVES*NQ)
#define LDS_FLOATS (2 * NSTATE + 3 * 256 + NWAVES * NQ + 16)

typedef float v2f __attribute__((ext_vector_type(2)));
typedef float v8f __attribute__((ext_vector_type(8)));

struct c32 { float x, y; };
__device__ __forceinline__ c32 cmul(c32 a, c32 b) {
  return c32{ a.x * b.x - a.y * b.y, a.x * b.y + a.y * b.x };
}
struct G2 { c32 m00, m01, m10, m11; };  // 2x2 complex gate
__device__ __forceinline__ c32 gsel(const G2& g, int rb, int cb) {
  c32 r0 = cb ? g.m01 : g.m00;
  c32 r1 = cb ? g.m11 : g.m10;
  return rb ? r1 : r0;
}
// offset of group-row-index r (4 bits) into the flat state index;
// slot j owns bit (3-j) of r and state-bit position p_j
__device__ __forceinline__ int roff(int r, int p0, int p1, int p2, int p3) {
  return (((r >> 3) & 1) << p0) | (((r >> 2) & 1) << p1) |
         (((r >> 1) & 1) << p2) | ((r & 1) << p3);
}
// insert a zero bit at (final) position pos
__device__ __forceinline__ int ins0(int v, int pos) {
  int m = (1 << pos) - 1;
  return (v & m) | ((v & ~m) << 1);
}

__global__ __launch_bounds__(BLOCK)
void HybridQuantumClassifier_qsim_kernel(const float* __restrict__ x,
                                         const float* __restrict__ wts,
                                         const float* __restrict__ fcw,
                                         const float* __restrict__ fcb,
                                         float* __restrict__ out) {
  extern __shared__ float lds[];
  float2* st  = (float2*)lds;            // interleaved state, NSTATE float2
  float*  Ure = lds + 2 * NSTATE;
  float*  Uim = Ure + 256;
  float*  Uin = Uim + 256;
  float*  red = Uin + 256;

  const int tid  = threadIdx.x;
  const int lane = tid & 31;
  const int wv   = tid >> 5;
  const int half = lane >> 4;      // 0: lanes 0-15, 1: lanes 16-31
  const int lm   = lane & 15;
  const int b    = blockIdx.x;

  // |0...0>
  for (int i = tid; i < NSTATE; i += BLOCK) st[i] = make_float2(0.0f, 0.0f);
  if (tid == 0) st[0] = make_float2(1.0f, 0.0f);
  __syncthreads();

  // stage 0 = AngleEmbedding (RX), stages 1..NL = StronglyEntanglingLayers
  for (int stage = 0; stage <= NL; ++stage) {
#pragma unroll
    for (int g = 0; g < 4; ++g) {
      // wires per group; group 3 pads with identity on wires 0,1 (already done)
      const int gwire0 = (g == 0) ? 0 : (g == 1) ? 4 : (g == 2) ? 8  : 12;
      const int gwire1 = (g == 0) ? 1 : (g == 1) ? 5 : (g == 2) ? 9  : 13;
      const int gwire2 = (g == 0) ? 2 : (g == 1) ? 6 : (g == 2) ? 10 : 0;
      const int gwire3 = (g == 0) ? 3 : (g == 1) ? 7 : (g == 2) ? 11 : 1;

      // Build the four 2x2 gates for this group (registers, per thread)
      G2 u[4];
#pragma unroll
      for (int j = 0; j < 4; ++j) {
        int wire = (j == 0) ? gwire0 : (j == 1) ? gwire1 : (j == 2) ? gwire2 : gwire3;
        bool act = (j < 2) || (g != 3);
        if (!act) {
          u[j].m00 = c32{1.f, 0.f}; u[j].m01 = c32{0.f, 0.f};
          u[j].m10 = c32{0.f, 0.f}; u[j].m11 = c32{1.f, 0.f};
        } else if (stage == 0) {
          float th = x[b * NQ + wire];
          float s, c; sincosf(0.5f * th, &s, &c);
          u[j].m00 = c32{c, 0.f};  u[j].m01 = c32{0.f, -s};
          u[j].m10 = c32{0.f, -s}; u[j].m11 = c32{c, 0.f};
        } else {
          const float* p = &wts[((stage - 1) * NQ + wire) * 3];
          float phi = p[0], th = p[1], om = p[2];
          float sh, ch; sincosf(0.5f * th, &sh, &ch);
          float sa, ca; sincosf(-0.5f * (phi + om), &sa, &ca);  // ep
          float sb, cb; sincosf( 0.5f * (phi - om), &sb, &cb);  // em
          u[j].m00 = c32{ ca * ch,  sa * ch};
          u[j].m01 = c32{-cb * sh, -sb * sh};
          u[j].m10 = c32{ cb * sh, -sb * sh};
          u[j].m11 = c32{ ca * ch, -sa * ch};
        }
      }

      // 16x16 kron product: one element per thread
      {
        int r = tid >> 4, c = tid & 15;
        c32 m = gsel(u[0], (r >> 3) & 1, (c >> 3) & 1);
        m = cmul(m, gsel(u[1], (r >> 2) & 1, (c >> 2) & 1));
        m = cmul(m, gsel(u[2], (r >> 1) & 1, (c >> 1) & 1));
        m = cmul(m, gsel(u[3], r & 1, c & 1));
        Ure[tid] = m.x; Uim[tid] = m.y; Uin[tid] = -m.y;
      }
      __syncthreads();

      const int p0 = 13 - gwire0, p1 = 13 - gwire1, p2 = 13 - gwire2, p3 = 13 - gwire3;
      // sorted ascending bit positions (constant-folds after unroll)
      int s0 = p0, s1 = p1, s2 = p2, s3 = p3, tswp;
      if (s0 > s1) { tswp = s0; s0 = s1; s1 = tswp; }
      if (s2 > s3) { tswp = s2; s2 = s3; s3 = tswp; }
      if (s0 > s2) { tswp = s0; s0 = s2; s2 = tswp; }
      if (s1 > s3) { tswp = s1; s1 = s3; s3 = tswp; }
      if (s1 > s2) { tswp = s1; s1 = s2; s2 = tswp; }

      // Hoisted A fragments (16x4 f32: lanes 0-15 K={4k,4k+1}, 16-31 K={4k+2,4k+3})
      v2f ar[4], ai[4], an[4];
#pragma unroll
      for (int k = 0; k < 4; ++k) {
        int c0 = 4 * k + 2 * half;
        ar[k] = v2f{ Ure[lm * 16 + c0], Ure[lm * 16 + c0 + 1] };
        ai[k] = v2f{ Uim[lm * 16 + c0], Uim[lm * 16 + c0 + 1] };
        an[k] = v2f{ Uin[lm * 16 + c0], Uin[lm * 16 + c0 + 1] };
      }

      // 64 independent 16-column tiles; 8 per wave (wave-uniform => EXEC all 1s)
      for (int t = wv; t < 64; t += NWAVES) {
        int col = t * 16 + lm;
        // scatter col's 10 bits around the 4 group bit positions (4 bit-inserts)
        int base = ins0(ins0(ins0(ins0(col, s0), s1), s2), s3);

        // B fragments (4x16 f32: lane = column, halves split K); b64 loads
        v2f br[4], bi[4];
#pragma unroll
        for (int k = 0; k < 4; ++k) {
          int r0 = 4 * k + 2 * half;
          float2 e0 = st[base + roff(r0,     p0, p1, p2, p3)];
          float2 e1 = st[base + roff(r0 + 1, p0, p1, p2, p3)];
          br[k] = v2f{ e0.x, e1.x };
          bi[k] = v2f{ e0.y, e1.y };
        }
        v8f accR = {0.f,0.f,0.f,0.f,0.f,0.f,0.f,0.f};
        v8f accI = {0.f,0.f,0.f,0.f,0.f,0.f,0.f,0.f};
        // Dr = Ur*Sr + (-Ui)*Si ; Di = Ur*Si + Ui*Sr
#pragma unroll
        for (int k = 0; k < 4; ++k)
          accR = __builtin_amdgcn_wmma_f32_16x16x4_f32(false, ar[k], false, br[k],
                                                       (short)0, accR, false, false);
#pragma unroll
        for (int k = 0; k < 4; ++k)
          accR = __builtin_amdgcn_wmma_f32_16x16x4_f32(false, an[k], false, bi[k],
                                                       (short)0, accR, false, false);
#pragma unroll
        for (int k = 0; k < 4; ++k)
          accI = __builtin_amdgcn_wmma_f32_16x16x4_f32(false, ar[k], false, bi[k],
                                                       (short)0, accI, false, false);
#pragma unroll
        for (int k = 0; k < 4; ++k)
          accI = __builtin_amdgcn_wmma_f32_16x16x4_f32(false, ai[k], false, br[k],
                                                       (short)0, accI, false, false);
        // D write-back (lane = N, VGPR v -> M = v + 8*half); b64 stores
#pragma unroll
        for (int v = 0; v < 8; ++v) {
          int m = v + 8 * half;
          st[base + roff(m, p0, p1, p2, p3)] = make_float2(accR[v], accI[v]);
        }
      }
      __syncthreads();
    }  // group

    // CNOT ring: ctrl w -> tgt (w + r) % NQ, r = (l mod 13) + 1
    if (stage >= 1) {
      int l = stage - 1;
      int rr = (l % (NQ - 1)) + 1;
      for (int wq = 0; wq < NQ; ++wq) {
        int tq = wq + rr; if (tq >= NQ) tq -= NQ;
        int pc = 13 - wq;
        int pt = 13 - tq;
        int lo = (pc < pt) ? pc : pt;
        int hi = (pc < pt) ? pt : pc;
        for (int m = tid; m < (NSTATE >> 2); m += BLOCK) {
          int a = ins0(ins0(m, lo), hi);   // zeros at pc and pt
          int i0 = a | (1 << pc);          // ctrl=1, tgt=0
          int i1 = i0 | (1 << pt);         // ctrl=1, tgt=1
          float2 t0 = st[i0];
          float2 t1 = st[i1];
          st[i0] = t1;
          st[i1] = t0;
        }
        __syncthreads();
      }
    }
  }  // stage

  // PauliZ expectation per wire + FC head
  float acc[NQ];
#pragma unroll
  for (int q = 0; q < NQ; ++q) acc[q] = 0.0f;
  for (int i = tid; i < NSTATE; i += BLOCK) {
    float2 a = st[i];
    float pr = a.x * a.x + a.y * a.y;
#pragma unroll
    for (int q = 0; q < NQ; ++q)
      acc[q] += ((i >> (13 - q)) & 1) ? -pr : pr;
  }
#pragma unroll
  for (int q = 0; q < NQ; ++q) {
    float v = acc[q];
    for (int off = 16; off > 0; off >>= 1) v += __shfl_down(v, off, 32);
    if (lane == 0) red[wv * NQ + q] = v;
  }
  __syncthreads();
  if (tid == 0) {
    float o = fcb[0];
    for (int q = 0; q < NQ; ++q) {
      float s = 0.0f;
      for (int k = 0; k < NWAVES; ++k) s += red[k * NQ + q];
      o += s * fcw[q];
    }
    out[b] = o;
  }
}

extern "C" void kernel_launch(void* const* d_in, const int* in_sizes, int n_in,
                              void* d_out, int out_size, void* d_ws, size_t ws_size,
                              hipStream_t stream) {
  const float* x   = (const float*)d_in[0];  // (512, 14)
  const float* wts = (const float*)d_in[1];  // (6, 14, 3)
  const float* fcw = (const float*)d_in[2];  // (1, 14)
  const float* fcb = (const float*)d_in[3];  // (1,)
  float* out = (float*)d_out;                // (512,)

  const int batch = in_sizes[0] / NQ;
  const size_t shmem = (size_t)LDS_FLOATS * sizeof(float);  // ~134 KB < 320 KB/WGP
  hipFuncSetAttribute(reinterpret_cast<const void*>(HybridQuantumClassifier_qsim_kernel),
                      hipFuncAttributeMaxDynamicSharedMemorySize, (int)shmem);
  HybridQuantumClassifier_qsim_kernel<<<batch, BLOCK, shmem, stream>>>(x, wts, fcw, fcb, out);
}